// EncoderLayer_43404939494207
// MI455X (gfx1250) — compile-verified
//
#include <hip/hip_runtime.h>
#include <math.h>
#include <stdint.h>

typedef __attribute__((ext_vector_type(2))) float v2f;
typedef __attribute__((ext_vector_type(8))) float v8f;
typedef __attribute__((ext_vector_type(4))) int v4i;

#define WMMA_F32(a, b, c) \
    __builtin_amdgcn_wmma_f32_16x16x4_f32(false, (a), false, (b), (short)0, (c), false, false)

// ---- CDNA5 async global->LDS copy path (guarded; sync fallback otherwise) ----
#if defined(__has_builtin)
#if __has_builtin(__builtin_amdgcn_global_load_async_to_lds_b128) && \
    __has_builtin(__builtin_amdgcn_s_wait_asynccnt)
#define USE_ASYNC_LDS 1
#endif
#endif

#ifdef USE_ASYNC_LDS
typedef __attribute__((address_space(1))) v4i* as1_v4i_ptr;
typedef __attribute__((address_space(3))) v4i* as3_v4i_ptr;
// inttoptr round-trip: bit-identical to addrspacecast on AMDGPU (AS1 = 64-bit VA;
// generic LDS-aperture address truncated to 32 bits == LDS offset, ISA 10.2)
#define CP128(gsrc, ldst)                                                      \
    __builtin_amdgcn_global_load_async_to_lds_b128(                            \
        (as1_v4i_ptr)(uintptr_t)(gsrc), (as3_v4i_ptr)(uintptr_t)(ldst), 0, 0)
#define WAIT_ASYNC(n) __builtin_amdgcn_s_wait_asynccnt(n)
#else
#define CP128(gsrc, ldst) \
    do { *(float4*)(ldst) = *(const float4*)(gsrc); } while (0)
#define WAIT_ASYNC(n) do {} while (0)
#endif

// ---------------------------------------------------------------------------
// LayerNorm (torch-style: std with n-1, eps added to std)
// ---------------------------------------------------------------------------
__global__ __launch_bounds__(256) void ln_kernel(const float* __restrict__ x,
                                                 const float* __restrict__ alpha,
                                                 const float* __restrict__ beta,
                                                 float* __restrict__ y) {
    const int D = 1024;
    const size_t row = blockIdx.x;
    const int t = threadIdx.x;
    const float4 xv = *(const float4*)(x + row * D + t * 4);

    __shared__ float red[8];
    float s = xv.x + xv.y + xv.z + xv.w;
    for (int off = 16; off >= 1; off >>= 1) s += __shfl_xor(s, off, 32);
    if ((t & 31) == 0) red[t >> 5] = s;
    __syncthreads();
    float tot = red[0] + red[1] + red[2] + red[3] + red[4] + red[5] + red[6] + red[7];
    const float mu = tot * (1.0f / 1024.0f);

    const float d0 = xv.x - mu, d1 = xv.y - mu, d2 = xv.z - mu, d3 = xv.w - mu;
    float s2 = d0 * d0 + d1 * d1 + d2 * d2 + d3 * d3;
    for (int off = 16; off >= 1; off >>= 1) s2 += __shfl_xor(s2, off, 32);
    __syncthreads();
    if ((t & 31) == 0) red[t >> 5] = s2;
    __syncthreads();
    float var = (red[0] + red[1] + red[2] + red[3] + red[4] + red[5] + red[6] + red[7]) *
                (1.0f / 1023.0f);
    const float inv = 1.0f / (sqrtf(var) + 1e-6f);

    const float4 av = *(const float4*)(alpha + t * 4);
    const float4 bv = *(const float4*)(beta + t * 4);
    float4 o;
    o.x = av.x * d0 * inv + bv.x;
    o.y = av.y * d1 * inv + bv.y;
    o.z = av.z * d2 * inv + bv.z;
    o.w = av.w * d3 * inv + bv.w;
    *(float4*)(y + row * D + t * 4) = o;
}

// ---------------------------------------------------------------------------
// fp32 WMMA GEMM:  C[M,N] = concat_K(A1[M,K1], A2[M,K2]) @ B[K,N] + bias
// optional ReLU / residual.  Block 128x128, BK=32, 8 waves (2Mx4N), each wave
// 64x32 = 4x2 accumulators.  Tiles staged with async global->LDS B128 copies,
// double-buffered: tile k+1 in flight while tile k feeds the WMMAs.
//   As row-major pad 4 (row 144B): A-frag ds_load_b64, conflict-free halves
//   Bs row-major pad 8 (row 544B): B-frag rows, conflict-free halves
// ---------------------------------------------------------------------------
template <bool RELU, bool RESID>
__global__ __launch_bounds__(256) void gemm_wmma(
    const float* __restrict__ A1, int lda1, int K1,
    const float* __restrict__ A2, int lda2, int K2,
    const float* __restrict__ Bmat, int ldb,
    const float* __restrict__ bias,
    const float* resid, float* C, int N) {
    __shared__ float As[2][128][36];
    __shared__ float Bs[2][32][136];

    const int bm = blockIdx.y * 128;
    const int bn = blockIdx.x * 128;
    const int t = threadIdx.x;
    const int wave = t >> 5;
    const int lane = t & 31;
    const bool hi = lane >= 16;
    const int l16 = lane & 15;
    const int wm = (wave >> 2) * 64;
    const int wn = (wave & 3) * 32;

    const int arow = t >> 3;         // 0..31
    const int ach = (t & 7) * 4;     // 0..28
    const int bk = t >> 5;           // 0..7
    const int bn4 = (t & 31) * 4;    // 0..124

    auto stage = [&](int k0, int buf) {
        const float* Asrc;
        int lda, kb;
        if (k0 < K1) { Asrc = A1; lda = lda1; kb = k0; }
        else         { Asrc = A2; lda = lda2; kb = k0 - K1; }
        for (int r = 0; r < 4; ++r) {
            const int m = r * 32 + arow;
            CP128(Asrc + (size_t)(bm + m) * lda + kb + ach, &As[buf][m][ach]);
        }
        for (int r = 0; r < 4; ++r) {
            const int kk = r * 8 + bk;
            CP128(Bmat + (size_t)(k0 + kk) * ldb + bn + bn4, &Bs[buf][kk][bn4]);
        }
    };

    const v8f zero8 = {0, 0, 0, 0, 0, 0, 0, 0};
    v8f acc[4][2];
    for (int i = 0; i < 4; ++i)
        for (int j = 0; j < 2; ++j) acc[i][j] = zero8;

    const int nTiles = (K1 + K2) >> 5;
    stage(0, 0);
    for (int it = 0; it < nTiles; ++it) {
        const int cur = it & 1;
        __syncthreads();  // (a) all waves done reading buffer cur^1
        if (it + 1 < nTiles) {
            stage((it + 1) * 32, cur ^ 1);
            WAIT_ASYNC(8);   // 8 async ops/wave per tile: tile `it` now complete
        } else {
            WAIT_ASYNC(0);
        }
        __syncthreads();  // (b) tile `it` visible to all waves

        const float(*Ac)[36] = As[cur];
        const float(*Bc)[136] = Bs[cur];
        for (int kk = 0; kk < 32; kk += 4) {
            const int ka = kk + (hi ? 2 : 0);
            v2f af[4], bf[2];
            for (int i = 0; i < 4; ++i) {
                const int m = wm + i * 16 + l16;
                af[i].x = Ac[m][ka + 0];
                af[i].y = Ac[m][ka + 1];
            }
            for (int j = 0; j < 2; ++j) {
                const int c = wn + j * 16 + l16;
                bf[j].x = Bc[ka + 0][c];
                bf[j].y = Bc[ka + 1][c];
            }
            for (int i = 0; i < 4; ++i)
                for (int j = 0; j < 2; ++j) acc[i][j] = WMMA_F32(af[i], bf[j], acc[i][j]);
        }
    }

    for (int i = 0; i < 4; ++i) {
        for (int j = 0; j < 2; ++j) {
            for (int r = 0; r < 8; ++r) {
                const int row = bm + wm + i * 16 + (hi ? 8 + r : r);
                const int col = bn + wn + j * 16 + l16;
                float val = acc[i][j][r] + bias[col];
                if (RESID) val += resid[(size_t)row * N + col];
                if (RELU) val = fmaxf(val, 0.0f);
                C[(size_t)row * N + col] = val;
            }
        }
    }
}

// ---------------------------------------------------------------------------
// Flash attention, fp32 WMMA.  S=2048, dh=64, row stride D=1024.
// grid: (B*H, S/64).  128 threads = 4 waves; each wave owns 16 q rows.
// K/V tiles double-buffered via async global->LDS; P transposed through LDS.
// ---------------------------------------------------------------------------
__global__ __launch_bounds__(128) void attn_kernel(const float* __restrict__ q,
                                                   const float* __restrict__ k,
                                                   const float* __restrict__ v,
                                                   float* __restrict__ ctx) {
    const int S = 2048, Dm = 1024, HD = 64;
    const int bh = blockIdx.x;
    const int b = bh >> 4, h = bh & 15;
    const int q0 = blockIdx.y * 64;
    const int t = threadIdx.x;
    const int wave = t >> 5;
    const int lane = t & 31;
    const bool hiL = lane >= 16;
    const int l16 = lane & 15;

    const float* qb = q + (size_t)b * S * Dm + h * HD;
    const float* kb = k + (size_t)b * S * Dm + h * HD;
    const float* vb = v + (size_t)b * S * Dm + h * HD;

    __shared__ float kt[2][64][68];   // pad 68 (272B rows): conflict-free reads
    __shared__ float vt[2][64][68];
    __shared__ float pb[4][16][65];   // per-wave P staging (C-layout -> A-layout)

    const int r0 = t >> 4;          // 0..7
    const int c4 = (t & 15) * 4;    // 0..60
    auto stageKV = [&](int kt0, int buf) {
        for (int it = 0; it < 8; ++it) {
            const int row = it * 8 + r0;
            CP128(kb + (size_t)(kt0 + row) * Dm + c4, &kt[buf][row][c4]);
            CP128(vb + (size_t)(kt0 + row) * Dm + c4, &vt[buf][row][c4]);
        }
    };

    // Q fragments (A layout), pre-scaled by 1/sqrt(64)
    v2f qf[16];
    {
        const int qrow = q0 + wave * 16 + l16;
        for (int ks = 0; ks < 16; ++ks) {
            const int kk = ks * 4 + (hiL ? 2 : 0);
            qf[ks].x = qb[(size_t)qrow * Dm + kk + 0] * 0.125f;
            qf[ks].y = qb[(size_t)qrow * Dm + kk + 1] * 0.125f;
        }
    }

    const v8f zero8 = {0, 0, 0, 0, 0, 0, 0, 0};
    v8f oacc[4];
    float mrow[8], lrow[8];
    for (int j = 0; j < 4; ++j) oacc[j] = zero8;
    for (int r = 0; r < 8; ++r) { mrow[r] = -1e30f; lrow[r] = 0.0f; }

    const int nTiles = S / 64;  // 32
    stageKV(0, 0);
    for (int it = 0; it < nTiles; ++it) {
        const int cur = it & 1;
        __syncthreads();  // (a) all waves done reading buffer cur^1
        if (it + 1 < nTiles) {
            stageKV((it + 1) * 64, cur ^ 1);
            WAIT_ASYNC(16);  // 16 async ops/wave per tile: tile `it` complete
        } else {
            WAIT_ASYNC(0);
        }
        __syncthreads();  // (b) tile `it` visible

        // S = Q K^T over dh=64  (B element [kk][key] = K[key][kk] from LDS)
        v8f sacc[4];
        for (int j = 0; j < 4; ++j) sacc[j] = zero8;
        for (int ks = 0; ks < 16; ++ks) {
            const int kk = ks * 4 + (hiL ? 2 : 0);
            const v2f af = qf[ks];
            for (int j = 0; j < 4; ++j) {
                v2f bf;
                bf.x = kt[cur][j * 16 + l16][kk + 0];
                bf.y = kt[cur][j * 16 + l16][kk + 1];
                sacc[j] = WMMA_F32(af, bf, sacc[j]);
            }
        }

        // online softmax: rows live across 16-lane halves, reduce via shfl_xor(16)
        for (int r = 0; r < 8; ++r) {
            float tm = fmaxf(fmaxf(sacc[0][r], sacc[1][r]), fmaxf(sacc[2][r], sacc[3][r]));
            for (int off = 8; off >= 1; off >>= 1) tm = fmaxf(tm, __shfl_xor(tm, off, 16));
            const float mn = fmaxf(mrow[r], tm);
            const float sc = __expf(mrow[r] - mn);
            mrow[r] = mn;
            float ts = 0.0f;
            for (int j = 0; j < 4; ++j) {
                const float pv = __expf(sacc[j][r] - mn);
                sacc[j][r] = pv;
                ts += pv;
            }
            for (int off = 8; off >= 1; off >>= 1) ts += __shfl_xor(ts, off, 16);
            lrow[r] = lrow[r] * sc + ts;
            for (int j = 0; j < 4; ++j) oacc[j][r] *= sc;
        }

        // stage P (C-layout regs -> row-major LDS)
        for (int j = 0; j < 4; ++j)
            for (int r = 0; r < 8; ++r)
                pb[wave][hiL ? r + 8 : r][j * 16 + l16] = sacc[j][r];
        __syncthreads();  // (c)

        // O += P @ V over keys (K-dim = 64)
        for (int ks = 0; ks < 16; ++ks) {
            const int kk = ks * 4 + (hiL ? 2 : 0);
            v2f af;
            af.x = pb[wave][l16][kk + 0];
            af.y = pb[wave][l16][kk + 1];
            for (int j = 0; j < 4; ++j) {
                v2f bf;
                bf.x = vt[cur][kk + 0][j * 16 + l16];
                bf.y = vt[cur][kk + 1][j * 16 + l16];
                oacc[j] = WMMA_F32(af, bf, oacc[j]);
            }
        }
    }

    // normalize and write ctx (B,S,H,dh)
    for (int j = 0; j < 4; ++j) {
        for (int r = 0; r < 8; ++r) {
            const int row = q0 + wave * 16 + (hiL ? r + 8 : r);
            const int col = h * HD + j * 16 + l16;
            ctx[((size_t)b * S + row) * Dm + col] = oacc[j][r] / lrow[r];
        }
    }
}

// ---------------------------------------------------------------------------
extern "C" void kernel_launch(void* const* d_in, const int* in_sizes, int n_in,
                              void* d_out, int out_size, void* d_ws, size_t ws_size,
                              hipStream_t stream) {
    (void)in_sizes; (void)n_in; (void)out_size; (void)ws_size;
    const float* feature_x = (const float*)d_in[0];
    const float* param_x   = (const float*)d_in[1];
    const float* Wq = (const float*)d_in[2];  const float* bq = (const float*)d_in[3];
    const float* Wk = (const float*)d_in[4];  const float* bk = (const float*)d_in[5];
    const float* Wv = (const float*)d_in[6];  const float* bv = (const float*)d_in[7];
    const float* Wo = (const float*)d_in[8];  const float* bo = (const float*)d_in[9];
    const float* alpha1 = (const float*)d_in[10]; const float* beta1 = (const float*)d_in[11];
    const float* alpha2 = (const float*)d_in[12]; const float* beta2 = (const float*)d_in[13];
    const float* W1 = (const float*)d_in[14]; const float* b1 = (const float*)d_in[15];
    const float* W2 = (const float*)d_in[16]; const float* b2 = (const float*)d_in[17];
    const float* Wp = (const float*)d_in[18]; const float* bp = (const float*)d_in[19];

    const size_t NE = (size_t)8192 * 1024;
    float* x = (float*)d_out;               // xn, then x (in-place residual epilogues)
    float* p = (float*)d_out + NE;          // (8192 x 256)
    float* ws = (float*)d_ws;
    float* qbuf = ws;
    float* kbuf = ws + NE;
    float* vbuf = ws + 2 * NE;
    float* cbuf = ws + 3 * NE;
    float* x2   = qbuf;  // reuse after attention
    float* hbuf = kbuf;  // reuse after attention (8192 x 512)

    // 1) xn = LN1(feature_x)
    ln_kernel<<<8192, 256, 0, stream>>>(feature_x, alpha1, beta1, x);

    // 2) Q/K/V projections (8192x1024 @ 1024x1024)
    gemm_wmma<false, false><<<dim3(8, 64), 256, 0, stream>>>(
        x, 1024, 1024, x, 0, 0, Wq, 1024, bq, nullptr, qbuf, 1024);
    gemm_wmma<false, false><<<dim3(8, 64), 256, 0, stream>>>(
        x, 1024, 1024, x, 0, 0, Wk, 1024, bk, nullptr, kbuf, 1024);
    gemm_wmma<false, false><<<dim3(8, 64), 256, 0, stream>>>(
        x, 1024, 1024, x, 0, 0, Wv, 1024, bv, nullptr, vbuf, 1024);

    // 3) flash attention -> ctx
    attn_kernel<<<dim3(64, 32), 128, 0, stream>>>(qbuf, kbuf, vbuf, cbuf);

    // 4) x = xn + ctx @ Wo + bo
    gemm_wmma<false, true><<<dim3(8, 64), 256, 0, stream>>>(
        cbuf, 1024, 1024, cbuf, 0, 0, Wo, 1024, bo, x, x, 1024);

    // 5) x2 = LN2(x)
    ln_kernel<<<8192, 256, 0, stream>>>(x, alpha2, beta2, x2);

    // 6) h = relu(x2 @ W1 + b1)
    gemm_wmma<true, false><<<dim3(4, 64), 256, 0, stream>>>(
        x2, 1024, 1024, x2, 0, 0, W1, 512, b1, nullptr, hbuf, 512);

    // 7) x = x + h @ W2 + b2
    gemm_wmma<false, true><<<dim3(8, 64), 256, 0, stream>>>(
        hbuf, 512, 512, hbuf, 0, 0, W2, 1024, b2, x, x, 1024);

    // 8) p = concat(x, param_x) @ Wp + bp  (split-K concat, 8192x1280 @ 1280x256)
    gemm_wmma<false, false><<<dim3(2, 64), 256, 0, stream>>>(
        x, 1024, 1024, param_x, 256, 256, Wp, 256, bp, nullptr, p, 256);
}